// EdgeConvLayer_28123445854493
// MI455X (gfx1250) — compile-verified
//
#include <hip/hip_runtime.h>

typedef __attribute__((ext_vector_type(2))) float v2f;
typedef __attribute__((ext_vector_type(8))) float v8f;

#define Bq 128
#define Nq 512
#define Fq 32
#define Hq 128
#define Dq 64
#define Kq 16

// ---------------------------------------------------------------------------
// Kernel 1: 2-D KNN (K=16) per point, within each batch of 512 points.
// One thread per point; batch coords staged in LDS; unrolled insertion sort.
// Strict '<' insertion preserves smaller-index-first on ties, matching the
// stable argsort of the reference. Self (j==i, dist 0) is skipped, matching
// the [1:1+K] slice.
// ---------------------------------------------------------------------------
__global__ __launch_bounds__(256) void knn_kernel(const float* __restrict__ events,
                                                  int* __restrict__ knn_idx) {
  __shared__ float sx[Nq];
  __shared__ float sy[Nq];
  const int tid = threadIdx.x;
  const int p   = blockIdx.x * 256 + tid;   // global point id, [0, B*N)
  const int b   = p >> 9;                   // 512 points per batch
  const int i   = p & (Nq - 1);

  for (int j = tid; j < Nq; j += 256) {
    const float* e = events + ((size_t)b * Nq + j) * Fq;
    sx[j] = e[0];
    sy[j] = e[1];
  }
  __syncthreads();

  const float cx = sx[i], cy = sy[i];
  float bestd[Kq];
  int   besti[Kq];
#pragma unroll
  for (int t = 0; t < Kq; ++t) { bestd[t] = 3.4e38f; besti[t] = 0; }

  for (int j = 0; j < Nq; ++j) {
    if (j == i) continue;
    const float dx = sx[j] - cx;
    const float dy = sy[j] - cy;
    const float d  = dx * dx + dy * dy;   // squared dist: same ordering as norm
    if (d < bestd[Kq - 1]) {
      float dd = d; int ii = j;
#pragma unroll
      for (int t = 0; t < Kq; ++t) {
        if (dd < bestd[t]) {
          const float td = bestd[t]; const int ti = besti[t];
          bestd[t] = dd; besti[t] = ii;
          dd = td; ii = ti;
        }
      }
    }
  }
#pragma unroll
  for (int t = 0; t < Kq; ++t) knn_idx[(size_t)p * Kq + t] = besti[t];
}

// ---------------------------------------------------------------------------
// Kernel 2: edge MLP via V_WMMA_F32_16X16X4_F32.
// One wave32 per point: M = 16 edges, layer1 K=64/N=128, layer2 K=128/N=64.
// A-operand f32 layout: lane L (m=L&15, h=L>>4), VGPR v holds E[m][4s+2h+v].
// B-operand: lane L, VGPR v holds W[4s+2h+v][16t + (L&15)].
// C/D layout: VGPR r holds out[r+8h][16t + (L&15)].
// ---------------------------------------------------------------------------
#define WAVES_PER_BLOCK 4
#define H_STRIDE 129   // 128 + 1 pad: kills LDS bank conflicts on transpose read

__global__ __launch_bounds__(WAVES_PER_BLOCK * 32) void edgeconv_mlp_kernel(
    const float* __restrict__ events, const int* __restrict__ knn_idx,
    const float* __restrict__ W1, const float* __restrict__ b1,
    const float* __restrict__ W2, const float* __restrict__ b2,
    float* __restrict__ out) {
  __shared__ float ldsH[WAVES_PER_BLOCK * 16 * H_STRIDE];

  const int lane = threadIdx.x & 31;
  const int wid  = threadIdx.x >> 5;
  const int p    = blockIdx.x * WAVES_PER_BLOCK + wid;  // point id
  const int b    = p >> 9;
  const int i    = p & (Nq - 1);
  const int m    = lane & 15;   // edge row owned by this lane
  const int h    = lane >> 4;   // K-half selector

  float* myH = ldsH + wid * 16 * H_STRIDE;

  const int nj = knn_idx[(size_t)p * Kq + m];
  const float* ev = events + ((size_t)b * Nq + i) * Fq;
  const float* nb = events + ((size_t)b * Nq + nj) * Fq;

  // Build this lane's half of edge row m directly in A-operand order:
  // a[2s+v] = E[m][col], col = 4s+2h+v (s<8: x part; s>=8: nbr-x part).
  float a[32];
#pragma unroll
  for (int s = 0; s < 16; ++s) {
    if (s < 8) {
      const int c = 4 * s + 2 * h;
      a[2 * s]     = ev[c];
      a[2 * s + 1] = ev[c + 1];
    } else {
      const int c = 4 * (s - 8) + 2 * h;
      a[2 * s]     = nb[c]     - ev[c];
      a[2 * s + 1] = nb[c + 1] - ev[c + 1];
    }
  }

  const v8f vzero = {};

  // ---- Layer 1: h1 = relu(E(16x64) @ W1(64x128) + b1) ----
  v8f acc1[8];
#pragma unroll
  for (int t = 0; t < 8; ++t) acc1[t] = vzero;

#pragma unroll
  for (int s = 0; s < 16; ++s) {
    v2f A;
    A.x = a[2 * s];
    A.y = a[2 * s + 1];
    const int kb = 4 * s + 2 * h;
    const float* w1r0 = W1 + (size_t)kb * Hq + m;
    const float* w1r1 = W1 + (size_t)(kb + 1) * Hq + m;
#pragma unroll
    for (int t = 0; t < 8; ++t) {
      v2f Bv;
      Bv.x = w1r0[16 * t];
      Bv.y = w1r1[16 * t];
      acc1[t] = __builtin_amdgcn_wmma_f32_16x16x4_f32(
          false, A, false, Bv, (short)0, acc1[t], false, false);
    }
  }

  // bias + relu, stash h1 (C-layout -> row-major LDS) for layer-2 transpose
#pragma unroll
  for (int t = 0; t < 8; ++t) {
    const float bv = b1[16 * t + m];
#pragma unroll
    for (int r = 0; r < 8; ++r) {
      float x = acc1[t][r] + bv;
      x = fmaxf(x, 0.0f);
      myH[(r + 8 * h) * H_STRIDE + 16 * t + m] = x;
    }
  }
  // same-wave LDS RAW: compiler inserts s_wait_dscnt; no barrier needed.

  // ---- Layer 2: h2 = relu(h1(16x128) @ W2(128x64) + b2) ----
  v8f acc2[4];
#pragma unroll
  for (int t = 0; t < 4; ++t) acc2[t] = vzero;

#pragma unroll
  for (int s = 0; s < 32; ++s) {
    const int kb = 4 * s + 2 * h;
    v2f A;
    A.x = myH[m * H_STRIDE + kb];
    A.y = myH[m * H_STRIDE + kb + 1];
    const float* w2r0 = W2 + (size_t)kb * Dq + m;
    const float* w2r1 = W2 + (size_t)(kb + 1) * Dq + m;
#pragma unroll
    for (int t = 0; t < 4; ++t) {
      v2f Bv;
      Bv.x = w2r0[16 * t];
      Bv.y = w2r1[16 * t];
      acc2[t] = __builtin_amdgcn_wmma_f32_16x16x4_f32(
          false, A, false, Bv, (short)0, acc2[t], false, false);
    }
  }

  // bias + relu + mean over the 16 edges (M dimension), then store.
#pragma unroll
  for (int t = 0; t < 4; ++t) {
    const float bv = b2[16 * t + m];
    float part = 0.0f;
#pragma unroll
    for (int r = 0; r < 8; ++r) part += fmaxf(acc2[t][r] + bv, 0.0f);
    part += __shfl_xor(part, 16, 32);   // combine M=0..7 and M=8..15 halves
    if (h == 0) out[(size_t)p * Dq + 16 * t + m] = part * (1.0f / 16.0f);
  }
}

// ---------------------------------------------------------------------------
extern "C" void kernel_launch(void* const* d_in, const int* in_sizes, int n_in,
                              void* d_out, int out_size, void* d_ws, size_t ws_size,
                              hipStream_t stream) {
  const float* events = (const float*)d_in[0];   // (128, 512, 32)
  const float* W1     = (const float*)d_in[1];   // (64, 128)
  const float* b1     = (const float*)d_in[2];   // (128,)
  const float* W2     = (const float*)d_in[3];   // (128, 64)
  const float* b2     = (const float*)d_in[4];   // (64,)
  float*       outp   = (float*)d_out;           // (128, 512, 64)
  int*         knn    = (int*)d_ws;              // 128*512*16 ints = 4 MB scratch

  knn_kernel<<<(Bq * Nq) / 256, 256, 0, stream>>>(events, knn);
  edgeconv_mlp_kernel<<<(Bq * Nq) / WAVES_PER_BLOCK, WAVES_PER_BLOCK * 32, 0, stream>>>(
      events, knn, W1, b1, W2, b2, outp);
}